// GroupEmbedding_72980084294355
// MI455X (gfx1250) — compile-verified
//
#include <hip/hip_runtime.h>
#include <stdint.h>

#define EMB 64
#define WAVES_PER_BLOCK 8

// ---------------------------------------------------------------------------
// Main kernel: one wave32 per user.
//   - u forced into an SGPR (readfirstlane) so the segment binary search and
//     all per-user/per-behavior index loads run on the scalar unit (s_load)
//   - double-buffered async gather of 256B item rows into LDS
//     (global_load_async_to_lds_b128, gated by s_wait_asynccnt)
//   - accumulate count-weighted rows (float2 per lane), scale by user row
//   - atomicAdd into out[group]  (output is 1MB -> L2-resident atomics)
// ---------------------------------------------------------------------------
__global__ __launch_bounds__(256) void ge_main_kernel(
    const int*   __restrict__ user_ids,
    const int*   __restrict__ user_group_seg,
    const int*   __restrict__ item_ids,
    const int*   __restrict__ behavior_user_seg,
    const float* __restrict__ counts,
    const float* __restrict__ user_table,
    const float* __restrict__ item_table,
    float*       __restrict__ out,
    int total_users, int total_beh)
{
  // per-wave double buffer: 2 x 64 floats = 512B per wave -> 4KB/block
  __shared__ float ldsbuf[WAVES_PER_BLOCK * 2 * EMB];

  const int lane = threadIdx.x & 31;
  const int wib  = threadIdx.x >> 5;               // wave index in block
  // wave-uniform user index, pinned to an SGPR so downstream address math,
  // branches and index loads scalarize (s_load / SALU, KMcnt)
  const int u = __builtin_amdgcn_readfirstlane(blockIdx.x * WAVES_PER_BLOCK + wib);
  if (u >= total_users) return;                    // scalar branch; no block sync used

  // ---- binary search: [start, end) = this user's behaviors (scalar loop) ----
  int lo = 0, hi = total_beh;
  while (lo < hi) {
    int mid = (lo + hi) >> 1;
    if (behavior_user_seg[mid] < u) lo = mid + 1; else hi = mid;
  }
  const int start = lo;
  hi = total_beh;
  while (lo < hi) {
    int mid = (lo + hi) >> 1;
    if (behavior_user_seg[mid] <= u) lo = mid + 1; else hi = mid;
  }
  const int end = lo;

  float* const buf0 = &ldsbuf[wib * 2 * EMB];
  float* const buf1 = buf0 + EMB;
  // generic->LDS: low 32 bits of a flat shared pointer are the LDS byte offset
  const unsigned off0 = (unsigned)(uintptr_t)buf0;
  const unsigned off1 = (unsigned)(uintptr_t)buf1;

  float acc0 = 0.0f, acc1 = 0.0f;

  // async-copy one 256B item row into LDS: 16 lanes x 16B (b128 each).
  // item_ids[b] is a scalar load; only the final address is per-lane.
  auto issue_row = [&](int b, unsigned ldsoff) {
    const int64_t row = (int64_t)item_ids[b] * EMB;
    const float*  gp  = item_table + row + lane * 4;   // 16B per lane
    const unsigned dst = ldsoff + (unsigned)(lane * 16);
    if (lane < 16) {
      asm volatile("global_load_async_to_lds_b128 %0, %1, off"
                   :: "v"(dst), "v"(gp)
                   : "memory");
    }
  };

  if (start < end) issue_row(start, off0);

  int parity = 0;
  for (int b = start; b < end; ++b) {
    const float c = counts[b];
    const bool more = (b + 1) < end;               // scalar condition
    if (more) {
      // prior ds_loads of the buffer we are about to overwrite must be done
      // (ASYNCcnt and DScnt are not cross-ordered)
      asm volatile("s_wait_dscnt 0" ::: "memory");
      issue_row(b + 1, parity ? off0 : off1);
      // async loads complete in order: cnt<=1 => current buffer is ready
      asm volatile("s_wait_asynccnt 1" ::: "memory");
    } else {
      asm volatile("s_wait_asynccnt 0" ::: "memory");
    }
    const float* cur = parity ? buf1 : buf0;
    acc0 = fmaf(c, cur[2 * lane + 0], acc0);       // ds_load_b64, bank-conflict-free
    acc1 = fmaf(c, cur[2 * lane + 1], acc1);
    parity ^= 1;
  }

  // personalize by user embedding, then reduce into the group slot
  const float* ut = user_table + (int64_t)user_ids[u] * EMB;
  const float p0 = acc0 * ut[2 * lane + 0];
  const float p1 = acc1 * ut[2 * lane + 1];

  float* op = out + (int64_t)user_group_seg[u] * EMB + 2 * lane;
  atomicAdd(op + 0, p0);
  atomicAdd(op + 1, p1);
}

// ---------------------------------------------------------------------------
// Kernel 1: zero the output (harness poisons d_out with 0xAA).
// ---------------------------------------------------------------------------
__global__ void ge_zero_kernel(float* __restrict__ out, int n) {
  int i = blockIdx.x * blockDim.x + threadIdx.x;
  if (i < n) out[i] = 0.0f;
}

// ---------------------------------------------------------------------------
// Host entry
// ---------------------------------------------------------------------------
extern "C" void kernel_launch(void* const* d_in, const int* in_sizes, int n_in,
                              void* d_out, int out_size, void* d_ws, size_t ws_size,
                              hipStream_t stream) {
  const int*   user_ids          = (const int*)d_in[0];
  const int*   user_group_seg    = (const int*)d_in[1];
  const int*   item_ids          = (const int*)d_in[2];
  const int*   behavior_user_seg = (const int*)d_in[3];
  const float* counts            = (const float*)d_in[4];
  const float* user_table        = (const float*)d_in[5];
  const float* item_table        = (const float*)d_in[6];
  float* out = (float*)d_out;

  const int total_users = in_sizes[0];
  const int total_beh   = in_sizes[2];

  ge_zero_kernel<<<(out_size + 255) / 256, 256, 0, stream>>>(out, out_size);

  const int grid = (total_users + WAVES_PER_BLOCK - 1) / WAVES_PER_BLOCK;
  ge_main_kernel<<<grid, 256, 0, stream>>>(
      user_ids, user_group_seg, item_ids, behavior_user_seg, counts,
      user_table, item_table, out, total_users, total_beh);

  (void)d_ws; (void)ws_size; (void)n_in;
}